// SimpleLiftNetwork_23313082483149
// MI455X (gfx1250) — compile-verified
//
#include <hip/hip_runtime.h>
#include <hip/hip_bf16.h>
#include <math.h>

#define D         64
#define KDIM      128          // 2*D
#define ROWS_PB   128          // rows per gemm block (4 waves x 32)

typedef __attribute__((ext_vector_type(2))) float v2f;
typedef __attribute__((ext_vector_type(8))) float v8f;

#if defined(__has_builtin)
#  if __has_builtin(__builtin_amdgcn_global_load_async_to_lds_b128) && \
      __has_builtin(__builtin_amdgcn_s_wait_asynccnt)
#    define USE_ASYNC_LDS 1
#  endif
#endif
#ifndef USE_ASYNC_LDS
#  define USE_ASYNC_LDS 0
#endif

#if USE_ASYNC_LDS
// exact parameter types per clang diagnostic: vector_size(16) int, AS1 / AS3
typedef int v4i_vs __attribute__((vector_size(16)));
typedef __attribute__((address_space(1))) v4i_vs* async_gptr_t;
typedef __attribute__((address_space(3))) v4i_vs* async_lptr_t;
#endif

// ---------------- zero aggr ----------------
__global__ void gnn_zero(float* __restrict__ p, int n4) {
    int i = blockIdx.x * blockDim.x + threadIdx.x;
    int stride = gridDim.x * blockDim.x;
    float4 z = make_float4(0.f, 0.f, 0.f, 0.f);
    for (; i < n4; i += stride) ((float4*)p)[i] = z;
}

// ---------------- edge scatter: aggr[dst] += x[src] * ew ----------------
// one wave per edge (grid-stride); lane handles columns 2*lane, 2*lane+1
__global__ void gnn_scatter(const float* __restrict__ x,
                            const long long* __restrict__ ei,
                            const float* __restrict__ ew,
                            float* __restrict__ aggr,
                            int nEdges) {
    int gid    = blockIdx.x * blockDim.x + threadIdx.x;
    int wave   = gid >> 5;
    int lane   = threadIdx.x & 31;
    int nWaves = (gridDim.x * blockDim.x) >> 5;
    for (int e = wave; e < nEdges; e += nWaves) {
        long long s = ei[e];
        long long d = ei[nEdges + e];
        float w = ew[e];
        float2 v = *(const float2*)(x + (size_t)s * D + lane * 2);
        float* ap = aggr + (size_t)d * D + lane * 2;
        atomicAdd(ap,     v.x * w);
        atomicAdd(ap + 1, v.y * w);
    }
}

// ---------------- fused: l2norm(aggr) | x  ->  GEMM (WMMA f32) -> bias -> row l2norm ----------------
// h = [l2norm(aggr) | xin] (ROWS_PB x 128) staged in LDS with XOR swizzle,
// xout = l2norm(h @ W + b).  Each wave owns a 32-row M tile (2 A-frags, 8 accs).
__global__ __launch_bounds__(128) void gnn_gemm_norm(
    const float* __restrict__ aggr, const float* __restrict__ xin,
    const float* __restrict__ Wg, const float* __restrict__ bg,
    float* __restrict__ xout, int nRows) {
    // element (r, k) stored at hlds[r*128 + (k ^ (4*(r&15)))]
    __shared__ float hlds[ROWS_PB * KDIM];

    const int t = threadIdx.x;
    const int rowBase = blockIdx.x * ROWS_PB;

    // ---- stage h rows rowBase..rowBase+127 into LDS (float4 granules) ----
    for (int i = t; i < ROWS_PB * (KDIM / 4); i += 128) {
        int r  = i >> 5;            // 32 float4 per 128-col row
        int c4 = (i & 31) * 4;
        int gr = rowBase + r;
        if (gr >= nRows) gr = nRows - 1;   // clamp; stores guarded later
        const float* gsrc = (c4 < D) ? (aggr + (size_t)gr * D + c4)
                                     : (xin  + (size_t)gr * D + (c4 - D));
        float* ldst = &hlds[r * KDIM + (c4 ^ ((r & 15) * 4))];
#if USE_ASYNC_LDS
        __builtin_amdgcn_global_load_async_to_lds_b128(
            (async_gptr_t)gsrc, (async_lptr_t)ldst, 0, 0);
#else
        *(float4*)ldst = *(const float4*)gsrc;
#endif
    }
#if USE_ASYNC_LDS
    __builtin_amdgcn_s_wait_asynccnt(0);
#endif
    __syncthreads();

    // ---- in-LDS l2 normalize of the aggr half (cols 0..63), one row per thread ----
    {
        const int r = t;                    // 128 rows, 128 threads
        const int swz = (r & 15) * 4;
        float* hr = &hlds[r * KDIM];
        float ssum = 0.f;
#pragma unroll
        for (int k = 0; k < D; k += 4) {
            float4 v = *(float4*)&hr[k ^ swz];
            ssum += v.x * v.x + v.y * v.y + v.z * v.z + v.w * v.w;
        }
        float inv = 1.0f / fmaxf(sqrtf(ssum), 1e-12f);
#pragma unroll
        for (int k = 0; k < D; k += 4) {
            float4 v = *(float4*)&hr[k ^ swz];
            v.x *= inv; v.y *= inv; v.z *= inv; v.w *= inv;
            *(float4*)&hr[k ^ swz] = v;
        }
    }
    __syncthreads();

    // ---- WMMA main loop ----
    const int wave  = t >> 5;
    const int lane  = t & 31;
    const int l15   = lane & 15;
    const int kh    = lane >> 4;        // half-wave select
    const int wrow0 = wave * 32;        // 32-row M tile per wave
    const int swz   = l15 * 4;

    v8f aA0 = {}, aA1 = {}, aA2 = {}, aA3 = {};   // rows wrow0..+15
    v8f aB0 = {}, aB1 = {}, aB2 = {}, aB3 = {};   // rows wrow0+16..+31

    const float* hA0 = &hlds[(wrow0 + l15) * KDIM];
    const float* hA1 = &hlds[(wrow0 + 16 + l15) * KDIM];

    for (int k0 = 0; k0 < KDIM; k0 += 4) {
        int kb  = k0 + 2 * kh;
        int off = kb ^ swz;             // even -> b64-aligned pair
        // A fragments (16x4 f32): lanes 0-15 -> K=k0,k0+1 ; lanes 16-31 -> K=k0+2,k0+3
        v2f a0 = { hA0[off], hA0[off + 1] };
        v2f a1 = { hA1[off], hA1[off + 1] };
        // B fragments (4x16 f32): b.x = W[kb][n], b.y = W[kb+1][n]
        const float* wk0 = Wg + (size_t)kb * D + l15;
        const float* wk1 = wk0 + D;
        v2f b0 = { wk0[0],  wk1[0]  };
        v2f b1 = { wk0[16], wk1[16] };
        v2f b2 = { wk0[32], wk1[32] };
        v2f b3 = { wk0[48], wk1[48] };
        aA0 = __builtin_amdgcn_wmma_f32_16x16x4_f32(false, a0, false, b0, (short)0, aA0, false, false);
        aA1 = __builtin_amdgcn_wmma_f32_16x16x4_f32(false, a0, false, b1, (short)0, aA1, false, false);
        aA2 = __builtin_amdgcn_wmma_f32_16x16x4_f32(false, a0, false, b2, (short)0, aA2, false, false);
        aA3 = __builtin_amdgcn_wmma_f32_16x16x4_f32(false, a0, false, b3, (short)0, aA3, false, false);
        aB0 = __builtin_amdgcn_wmma_f32_16x16x4_f32(false, a1, false, b0, (short)0, aB0, false, false);
        aB1 = __builtin_amdgcn_wmma_f32_16x16x4_f32(false, a1, false, b1, (short)0, aB1, false, false);
        aB2 = __builtin_amdgcn_wmma_f32_16x16x4_f32(false, a1, false, b2, (short)0, aB2, false, false);
        aB3 = __builtin_amdgcn_wmma_f32_16x16x4_f32(false, a1, false, b3, (short)0, aB3, false, false);
    }

    // ---- bias + per-row l2 norm + store (two 16-row sets) ----
    float bn0 = bg[l15], bn1 = bg[16 + l15], bn2 = bg[32 + l15], bn3 = bg[48 + l15];

    float ssA[8], ssB[8];
#pragma unroll
    for (int j = 0; j < 8; ++j) {
        float c0 = aA0[j] + bn0, c1 = aA1[j] + bn1, c2 = aA2[j] + bn2, c3 = aA3[j] + bn3;
        aA0[j] = c0; aA1[j] = c1; aA2[j] = c2; aA3[j] = c3;
        ssA[j] = c0 * c0 + c1 * c1 + c2 * c2 + c3 * c3;
        float d0 = aB0[j] + bn0, d1 = aB1[j] + bn1, d2 = aB2[j] + bn2, d3 = aB3[j] + bn3;
        aB0[j] = d0; aB1[j] = d1; aB2[j] = d2; aB3[j] = d3;
        ssB[j] = d0 * d0 + d1 * d1 + d2 * d2 + d3 * d3;
    }

#pragma unroll
    for (int j = 0; j < 8; ++j) {
        // C/D layout: VGPR j holds rows j (lanes 0-15) and j+8 (lanes 16-31)
        float sA = ssA[j];
        sA += __shfl_xor(sA, 1, 32);
        sA += __shfl_xor(sA, 2, 32);
        sA += __shfl_xor(sA, 4, 32);
        sA += __shfl_xor(sA, 8, 32);
        float invA = 1.0f / fmaxf(sqrtf(sA), 1e-12f);
        int rowA = rowBase + wrow0 + j + 8 * kh;
        if (rowA < nRows) {
            float* o = xout + (size_t)rowA * D + l15;
            o[0]  = aA0[j] * invA;
            o[16] = aA1[j] * invA;
            o[32] = aA2[j] * invA;
            o[48] = aA3[j] * invA;
        }
        float sB = ssB[j];
        sB += __shfl_xor(sB, 1, 32);
        sB += __shfl_xor(sB, 2, 32);
        sB += __shfl_xor(sB, 4, 32);
        sB += __shfl_xor(sB, 8, 32);
        float invB = 1.0f / fmaxf(sqrtf(sB), 1e-12f);
        int rowB = rowA + 16;
        if (rowB < nRows) {
            float* o = xout + (size_t)rowB * D + l15;
            o[0]  = aB0[j] * invB;
            o[16] = aB1[j] * invB;
            o[32] = aB2[j] * invB;
            o[48] = aB3[j] * invB;
        }
    }
}

extern "C" void kernel_launch(void* const* d_in, const int* in_sizes, int n_in,
                              void* d_out, int out_size, void* d_ws, size_t ws_size,
                              hipStream_t stream) {
    const float*      x_in = (const float*)d_in[0];
    const long long*  ei   = (const long long*)d_in[1];
    const float*      ew   = (const float*)d_in[2];
    const float*      Wg   = (const float*)d_in[3];
    const float*      bg   = (const float*)d_in[4];

    const int nNodes  = in_sizes[0] / D;
    const int nEdges  = in_sizes[2];
    const int nLayers = in_sizes[4] / D;

    // workspace: [aggr | xbuf0 | xbuf1]
    const size_t nd = (size_t)nNodes * D;
    float* aggr = (float*)d_ws;
    float* buf0 = aggr + nd;
    float* buf1 = buf0 + nd;

    const int zeroBlocks    = 1024;
    const int scatterBlocks = 1280;
    const int gemmBlocks    = (nNodes + ROWS_PB - 1) / ROWS_PB;

    const float* xc = x_in;
    for (int L = 0; L < nLayers; ++L) {
        float* xo = (L == nLayers - 1) ? (float*)d_out
                                       : ((L & 1) == 0 ? buf0 : buf1);
        gnn_zero<<<zeroBlocks, 256, 0, stream>>>(aggr, (int)(nd / 4));
        gnn_scatter<<<scatterBlocks, 256, 0, stream>>>(xc, ei, ew, aggr, nEdges);
        gnn_gemm_norm<<<gemmBlocks, 128, 0, stream>>>(
            aggr, xc, Wg + (size_t)L * KDIM * D, bg + (size_t)L * D, xo, nNodes);
        xc = xo;
    }
}